// SiameseGNN_43559558316703
// MI455X (gfx1250) — compile-verified
//
#include <hip/hip_runtime.h>
#include <math.h>

// ---------------------------------------------------------------------------
// Siamese 2-layer GCN + projection head + null head, tuned for MI455X
// (gfx1250, wave32, WMMA). Edge aggregation is L2-resident scatter/gather;
// dense GEMMs use V_WMMA_F32_16X16X4_F32 (full f32 precision).
// LDS weight panel is K-pair interleaved so each B fragment is one
// aligned ds_load_b64; A/B pointers are strength-reduced running pointers
// and the k-loop is force-unrolled 4x for a 16-WMMA pipelined body.
// ---------------------------------------------------------------------------

#define NN 50000
#define EE 800000

typedef float v2f __attribute__((ext_vector_type(2)));
typedef float v8f __attribute__((ext_vector_type(8)));

__device__ __forceinline__ void atomAddF(float* p, float v) {
  (void)__hip_atomic_fetch_add(p, v, __ATOMIC_RELAXED, __HIP_MEMORY_SCOPE_AGENT);
}

// ---------------- tiny utility kernels ----------------
__global__ void k_zero(float* __restrict__ p, int n) {
  int i = blockIdx.x * blockDim.x + threadIdx.x;
  if (i < n) p[i] = 0.0f;
}

__global__ void k_deg(const int* __restrict__ dst, float* __restrict__ deg, int E) {
  int e = blockIdx.x * blockDim.x + threadIdx.x;
  if (e < E) atomAddF(&deg[dst[e]], 1.0f);
}

// in-place: deg -> rsqrt(deg + 1)
__global__ void k_norm(float* __restrict__ d, int n) {
  int i = blockIdx.x * blockDim.x + threadIdx.x;
  if (i < n) d[i] = rsqrtf(d[i] + 1.0f);
}

// agg[v] = x[v] * norm[v]^2   (self-loop term; also zero-initializes agg)
__global__ void k_selfinit(const float* __restrict__ x, const float* __restrict__ nrm,
                           float* __restrict__ agg) {
  int i = blockIdx.x * blockDim.x + threadIdx.x;  // float4 index, N*32 total
  int node = i >> 5;
  float nn = nrm[node];
  nn = nn * nn;
  float4 v = ((const float4*)x)[i];
  v.x *= nn; v.y *= nn; v.z *= nn; v.w *= nn;
  ((float4*)agg)[i] = v;
}

// One wave32 per edge: 32 lanes x float4 == one 512B feature row.
// agg[dst] += x[src] * norm[src]*norm[dst]   (L2-resident atomics)
__global__ void k_scatter(const float* __restrict__ x, const int* __restrict__ src,
                          const int* __restrict__ dst, const float* __restrict__ nrm,
                          float* __restrict__ agg, int E) {
  int lane = threadIdx.x & 31;
  int e = blockIdx.x * 8 + (threadIdx.x >> 5);
  if (e >= E) return;
  int s = src[e];
  int d = dst[e];
  if (lane == 0 && e + 8 < E) {
    // hint the next edge's source row toward the caches (global_prefetch_b8)
    __builtin_prefetch(x + (size_t)src[e + 8] * 128, 0, 1);
  }
  float w = nrm[s] * nrm[d];
  float4 v = ((const float4*)x)[(size_t)s * 32 + lane];
  float* a = agg + (size_t)d * 128 + lane * 4;
  atomAddF(a + 0, v.x * w);
  atomAddF(a + 1, v.y * w);
  atomAddF(a + 2, v.z * w);
  atomAddF(a + 3, v.w * w);
}

// ---------------- WMMA GEMM: out = act(A @ W + b) ----------------
// A = [A0 (K0 wide) | A1 (K1 wide)] row-major, W row-major [Ktot][wStride],
// out row-major [nRows][outStride]. Block = 8 waves; wave -> 16 rows x 64 cols.
// blockIdx.y selects a 64-column panel.
//
// LDS layout: K-pair interleaved float2 rows:
//   lw[(k>>1)*PAIR_STRIDE + 2*n + (k&1)] = W[k][cb+n]
// so B fragment (W[kA][n], W[kA+1][n]) with even kA is a single aligned
// ds_load_b64. PAIR_STRIDE=160 dwords -> rows p and p+1 hit disjoint
// 32-bank halves (160 % 64 == 32): conflict-free for the two half-waves.
#define PAIR_STRIDE 160

#define WMMA_STEP(aptr, bptr)                                                          \
  do {                                                                                 \
    v2f a = *(const v2f*)(aptr);                                                       \
    v2f b0 = *(const v2f*)((bptr) + 0);                                                \
    v2f b1 = *(const v2f*)((bptr) + 32);                                               \
    v2f b2 = *(const v2f*)((bptr) + 64);                                               \
    v2f b3 = *(const v2f*)((bptr) + 96);                                               \
    acc0 = __builtin_amdgcn_wmma_f32_16x16x4_f32(false, a, false, b0, (short)0, acc0,  \
                                                 false, false);                        \
    acc1 = __builtin_amdgcn_wmma_f32_16x16x4_f32(false, a, false, b1, (short)0, acc1,  \
                                                 false, false);                        \
    acc2 = __builtin_amdgcn_wmma_f32_16x16x4_f32(false, a, false, b2, (short)0, acc2,  \
                                                 false, false);                        \
    acc3 = __builtin_amdgcn_wmma_f32_16x16x4_f32(false, a, false, b3, (short)0, acc3,  \
                                                 false, false);                        \
  } while (0)

__global__ void k_gemm(const float* __restrict__ A0, int K0,
                       const float* __restrict__ A1, int K1,
                       const float* __restrict__ W, int wStride,
                       const float* __restrict__ bias,
                       float* __restrict__ out, int outStride,
                       int Ktot, int doRelu, int nRows) {
  __shared__ __align__(16) float lw[96 * PAIR_STRIDE];  // up to K=192, 61.4 KB

  const int tid = threadIdx.x;
  const int cb = blockIdx.y * 64;

  // cooperative load of the 64-wide weight panel into LDS (pair-interleaved)
  for (int i = tid; i < Ktot * 64; i += 256) {
    int k = i >> 6;
    int n = i & 63;
    lw[(k >> 1) * PAIR_STRIDE + 2 * n + (k & 1)] = W[(size_t)k * wStride + cb + n];
  }
  __syncthreads();

  const int wave = tid >> 5;
  const int lane = tid & 31;
  const int rowTile = blockIdx.x * 8 + wave;
  if (rowTile * 16 >= nRows) return;  // wave-uniform: EXEC stays all-ones

  const int half = lane >> 4;   // 0: lanes 0-15, 1: lanes 16-31
  const int mi   = lane & 15;
  const int rowA = rowTile * 16 + mi;

  v8f acc0 = {}, acc1 = {}, acc2 = {}, acc3 = {};

  // Running pointers: A advances 4 floats / step, B advances one K-pair-pair
  // (2*PAIR_STRIDE floats) / step -- both fold into instruction offsets.
  const float* aP = A0 + (size_t)rowA * K0 + half * 2;
  const float* bP = lw + half * PAIR_STRIDE + 2 * mi;

  const int steps0 = K0 >> 2;
#pragma unroll 4
  for (int s = 0; s < steps0; ++s) {
    WMMA_STEP(aP, bP);
    aP += 4;
    bP += 2 * PAIR_STRIDE;
  }

  const int steps1 = (Ktot - K0) >> 2;
  if (steps1 > 0) {
    aP = A1 + (size_t)rowA * K1 + half * 2;
#pragma unroll 4
    for (int s = 0; s < steps1; ++s) {
      WMMA_STEP(aP, bP);
      aP += 4;
      bP += 2 * PAIR_STRIDE;
    }
  }

  // epilogue: bias + optional ReLU, scatter C tiles to global
  v8f accs[4] = {acc0, acc1, acc2, acc3};
#pragma unroll
  for (int nt = 0; nt < 4; ++nt) {
    int colg = cb + nt * 16 + mi;
    float bv = bias[colg];
#pragma unroll
    for (int j = 0; j < 8; ++j) {
      int rowO = rowTile * 16 + j + half * 8;  // C layout: VGPR j -> M=j (+8 for hi half)
      float v = accs[nt][j] + bv;
      if (doRelu) v = fmaxf(v, 0.0f);
      out[(size_t)rowO * outStride + colg] = v;
    }
  }
}

// ---------------- row-wise wave32 reductions ----------------
// in-place L2 normalize rows of z [N,64]; one wave per row, 8 rows/block
__global__ void k_l2norm(float* __restrict__ z) {
  int lane = threadIdx.x & 31;
  int row = blockIdx.x * 8 + (threadIdx.x >> 5);
  float a = z[(size_t)row * 64 + lane];
  float b = z[(size_t)row * 64 + lane + 32];
  float s = a * a + b * b;
#pragma unroll
  for (int off = 16; off > 0; off >>= 1) s += __shfl_xor(s, off, 32);
  float inv = 1.0f / fmaxf(sqrtf(s), 1e-12f);
  z[(size_t)row * 64 + lane]      = a * inv;
  z[(size_t)row * 64 + lane + 32] = b * inv;
}

// out[row] = u[row,:] . w + b   (u: [N,64], w: [64], b: scalar)
__global__ void k_dot64(const float* __restrict__ u, const float* __restrict__ w,
                        const float* __restrict__ b, float* __restrict__ out) {
  int lane = threadIdx.x & 31;
  int row = blockIdx.x * 8 + (threadIdx.x >> 5);
  float s = u[(size_t)row * 64 + lane] * w[lane] +
            u[(size_t)row * 64 + lane + 32] * w[lane + 32];
#pragma unroll
  for (int off = 16; off > 0; off >>= 1) s += __shfl_xor(s, off, 32);
  if (lane == 0) out[row] = s + b[0];
}

// ---------------------------------------------------------------------------
extern "C" void kernel_launch(void* const* d_in, const int* in_sizes, int n_in,
                              void* d_out, int out_size, void* d_ws, size_t ws_size,
                              hipStream_t stream) {
  (void)in_sizes; (void)n_in; (void)out_size; (void)ws_size;
  const size_t N = NN;
  const int E = EE;

  const float* xA  = (const float*)d_in[0];
  const float* xB  = (const float*)d_in[1];
  const int*   eiA = (const int*)d_in[2];
  const int*   eiB = (const int*)d_in[3];
  const float* W1  = (const float*)d_in[4];
  const float* b1  = (const float*)d_in[5];
  const float* W2  = (const float*)d_in[6];
  const float* b2  = (const float*)d_in[7];
  const float* Wp1 = (const float*)d_in[8];
  const float* bp1 = (const float*)d_in[9];
  const float* Wp2 = (const float*)d_in[10];
  const float* bp2 = (const float*)d_in[11];
  const float* Wn1 = (const float*)d_in[12];
  const float* bn1 = (const float*)d_in[13];
  const float* Wn2 = (const float*)d_in[14];
  const float* bn2 = (const float*)d_in[15];

  // workspace layout (floats)
  float* ws   = (float*)d_ws;
  float* nrm  = ws;                  // N
  float* agg  = nrm + N;             // N*128
  float* h    = agg + N * 128;       // N*128
  float* b64  = h + N * 128;         // N*64

  // output layout (floats): hA, hB, zA, zB, nullA, nullB
  float* out = (float*)d_out;
  float* hA = out;
  float* hB = out + N * 128;
  float* zA = out + 2 * N * 128;
  float* zB = zA + N * 64;
  float* nA = zB + N * 64;
  float* nB = nA + N;

  const dim3 blk(256);
  const dim3 gN((NN + 255) / 256);       // 196
  const dim3 gE(E / 256);                // 3125
  const dim3 gSelf(NN * 32 / 256);       // 6250
  const dim3 gScat(E / 8);               // 100000
  const dim3 gRow(NN / 8);               // 6250
  const dim3 gGemm128((NN / 16 + 7) / 8, 2);  // (391, 2): 128-col output
  const dim3 gGemm64((NN / 16 + 7) / 8, 1);   // (391, 1): 64-col output

  auto run_graph = [&](const float* x, const int* ei, float* hOut, float* zOut,
                       float* nOut) {
    const int* src = ei;
    const int* dst = ei + E;

    // degree -> norm = rsqrt(deg + 1)
    k_zero<<<gN, blk, 0, stream>>>(nrm, NN);
    k_deg<<<gE, blk, 0, stream>>>(dst, nrm, E);
    k_norm<<<gN, blk, 0, stream>>>(nrm, NN);

    // GCN layer 1: agg = self + scatter(x) ; h = relu(agg @ W1 + b1)
    k_selfinit<<<gSelf, blk, 0, stream>>>(x, nrm, agg);
    k_scatter<<<gScat, blk, 0, stream>>>(x, src, dst, nrm, agg, E);
    k_gemm<<<gGemm128, blk, 0, stream>>>(agg, 128, agg, 0, W1, 128, b1,
                                         h, 128, 128, 1, NN);

    // GCN layer 2: agg = self + scatter(h) ; hOut = agg @ W2 + b2
    k_selfinit<<<gSelf, blk, 0, stream>>>(h, nrm, agg);
    k_scatter<<<gScat, blk, 0, stream>>>(h, src, dst, nrm, agg, E);
    k_gemm<<<gGemm128, blk, 0, stream>>>(agg, 128, agg, 0, W2, 128, b2,
                                         hOut, 128, 128, 0, NN);

    // projection head: z = l2norm(relu(h @ Wp1 + bp1) @ Wp2 + bp2)
    k_gemm<<<gGemm64, blk, 0, stream>>>(hOut, 128, hOut, 0, Wp1, 64, bp1,
                                        b64, 64, 128, 1, NN);
    k_gemm<<<gGemm64, blk, 0, stream>>>(b64, 64, b64, 0, Wp2, 64, bp2,
                                        zOut, 64, 64, 0, NN);
    k_l2norm<<<gRow, blk, 0, stream>>>(zOut);

    // null head on concat [h | z] (dual-source A, Ktot = 192)
    k_gemm<<<gGemm64, blk, 0, stream>>>(hOut, 128, zOut, 64, Wn1, 64, bn1,
                                        b64, 64, 192, 1, NN);
    k_dot64<<<gRow, blk, 0, stream>>>(b64, Wn2, bn2, nOut);
  };

  run_graph(xA, eiA, hA, zA, nA);
  run_graph(xB, eiB, hB, zB, nB);
}